// AdjModel_22436909154811
// MI455X (gfx1250) — compile-verified
//
#include <hip/hip_runtime.h>

// ---------------- problem constants (match reference) ----------------
#define Bn 4
#define Nn 5000
#define Kn 16
#define Ln 3
#define Fn 2
#define En 32
#define Dn 64
#define NKc (Nn*Kn)
#define BIGC 1000000000.0f

typedef __attribute__((ext_vector_type(2))) float v2f;
typedef __attribute__((ext_vector_type(8))) float v8f;

// D(16x16,f32) = A(16x4,f32) x B(4x16,f32) + C  -- fp32 matrix path, keeps
// reference numerics exactly while using the CDNA5 WMMA pipe.
__device__ __forceinline__ v8f wmma4(v2f a, v2f b, v8f c) {
  return __builtin_amdgcn_wmma_f32_16x16x4_f32(false, a, false, b, (short)0, c,
                                               false, false);
}

// ---------------- encoder: h = [emb_norm, feat] @ enc_W + enc_b -------------
__global__ __launch_bounds__(64) void k_encoder(
    const float* __restrict__ embt, const float* __restrict__ feat,
    const float* __restrict__ W, const float* __restrict__ bias,
    float* __restrict__ h) {
  __shared__ float in[En + Fn];
  int node = blockIdx.x;            // b*N + n
  int n = node % Nn;
  int tid = threadIdx.x;
  if (tid < En) {
    float v = embt[n * En + tid];
    float ss = v * v;
    for (int o = 16; o; o >>= 1) ss += __shfl_xor(ss, o, 32);
    float nr = sqrtf(ss);
    in[tid] = v / fmaxf(nr, 1.0f);
  } else if (tid < En + Fn) {
    in[tid] = feat[node * Fn + (tid - En)];
  }
  __syncthreads();
  float acc = bias[tid];
#pragma unroll
  for (int k = 0; k < En + Fn; ++k) acc += in[k] * W[k * Dn + tid];
  h[node * Dn + tid] = acc;
}

// ---------------- neighborhood mean aggregation ----------------
__global__ __launch_bounds__(64) void k_agg(
    const float* __restrict__ h, const int* __restrict__ neigh,
    const int* __restrict__ num_nodes, float* __restrict__ agg) {
  __shared__ int idxs[Kn];
  int r = blockIdx.x;               // (b*N+n)*L + t
  int t = r % Ln;
  int bn = r / Ln;
  int n = bn % Nn, b = bn / Nn;
  int tid = threadIdx.x;
  int padv = num_nodes[b];
  if (tid < Kn) idxs[tid] = neigh[((t * Bn + b) * Nn + n) * Kn + tid];
  __syncthreads();
  float s = 0.f;
  int deg = 0;
#pragma unroll
  for (int k = 0; k < Kn; ++k) {
    int idx = idxs[k];
    if (idx != padv) {
      int ic = idx < Nn ? idx : Nn - 1;
      s += h[(b * Nn + ic) * Dn + tid];
      deg++;
    }
  }
  agg[r * Dn + tid] = s / fmaxf((float)deg, 1.0f);
}

// ---- LDS weight staging in k-pair-interleaved layout -----------------------
// wp[(k/2)*2*NCOL + col*2 + (k&1)] = W[k*NCOL + col]
// -> B fragment {W[k][col], W[k+1][col]} is one aligned 8-byte ds_load_b64.
template <int NCOL, int KDIM>
__device__ __forceinline__ void stage_pairs(const float* __restrict__ W,
                                            float* __restrict__ wp, int tid,
                                            int nthreads) {
  for (int i = tid; i < KDIM * NCOL; i += nthreads) {
    int k = i / NCOL, col = i - k * NCOL;
    wp[(k >> 1) * (2 * NCOL) + col * 2 + (k & 1)] = W[i];
  }
}

template <int NCOL>
__device__ __forceinline__ v2f bfrag(const float* __restrict__ wp, int k0,
                                     int col) {
  // k0 even; returns {W[k0][col], W[k0+1][col]}
  return ((const v2f*)wp)[(k0 >> 1) * NCOL + col];
}

// ---------------- generic (optionally fused dual) M x 64 @ 64 x 64 GEMM -----
// ACT: 0 none, 1 tanh, 2 sigmoid.  DUAL: C = act(A1@W1 [+ (Mul.*A2)@W2] + b)
template <int ACT>
__device__ __forceinline__ void gemm_store(v8f acc, const float* bias, float* C,
                                           int row0, int half, int l16,
                                           int ct) {
  int col = ct * 16 + l16;
  float bv = bias ? bias[col] : 0.f;
#pragma unroll
  for (int v = 0; v < 8; ++v) {
    float val = acc[v] + bv;
    if (ACT == 1) val = tanhf(val);
    else if (ACT == 2) val = 1.f / (1.f + __expf(-val));
    C[(row0 + half * 8 + v) * Dn + col] = val;
  }
}

template <int ACT, bool DUAL, bool HASMUL>
__global__ __launch_bounds__(256) void k_gemm64(
    const float* __restrict__ A1, const float* __restrict__ W1,
    const float* __restrict__ A2, const float* __restrict__ W2,
    const float* __restrict__ Mul, const float* __restrict__ bias,
    float* __restrict__ C, int M) {
  __shared__ float w1s[Dn * Dn];
  __shared__ float w2s[DUAL ? Dn * Dn : 1];
  int tid = threadIdx.x, lane = tid & 31, wave = tid >> 5;
  stage_pairs<Dn, Dn>(W1, w1s, tid, 256);
  if (DUAL) stage_pairs<Dn, Dn>(W2, w2s, tid, 256);
  __syncthreads();
  int tile = blockIdx.x * 8 + wave;
  int row0 = tile * 16;
  if (row0 >= M) return;                 // wave-uniform, EXEC stays full
  int half = lane >> 4, l16 = lane & 15;
  int arow = row0 + l16;
  v8f acc0 = {}, acc1 = {}, acc2 = {}, acc3 = {};
#pragma unroll 4
  for (int kt = 0; kt < 16; ++kt) {
    int k0 = kt * 4 + half * 2;
    v2f a = *(const v2f*)(A1 + arow * Dn + k0);
    acc0 = wmma4(a, bfrag<Dn>(w1s, k0, l16), acc0);
    acc1 = wmma4(a, bfrag<Dn>(w1s, k0, 16 + l16), acc1);
    acc2 = wmma4(a, bfrag<Dn>(w1s, k0, 32 + l16), acc2);
    acc3 = wmma4(a, bfrag<Dn>(w1s, k0, 48 + l16), acc3);
    if (DUAL) {
      v2f a2 = *(const v2f*)(A2 + arow * Dn + k0);
      if (HASMUL) a2 = a2 * (*(const v2f*)(Mul + arow * Dn + k0));
      acc0 = wmma4(a2, bfrag<Dn>(w2s, k0, l16), acc0);
      acc1 = wmma4(a2, bfrag<Dn>(w2s, k0, 16 + l16), acc1);
      acc2 = wmma4(a2, bfrag<Dn>(w2s, k0, 32 + l16), acc2);
      acc3 = wmma4(a2, bfrag<Dn>(w2s, k0, 48 + l16), acc3);
    }
  }
  gemm_store<ACT>(acc0, bias, C, row0, half, l16, 0);
  gemm_store<ACT>(acc1, bias, C, row0, half, l16, 1);
  gemm_store<ACT>(acc2, bias, C, row0, half, l16, 2);
  gemm_store<ACT>(acc3, bias, C, row0, half, l16, 3);
}

// ---------------- per-node attention (H=4 heads, L=3 slots) ----------------
__global__ __launch_bounds__(256) void k_attn(
    const float* __restrict__ q, const float* __restrict__ kk,
    const float* __restrict__ vv, float* __restrict__ outp) {
  int i = blockIdx.x * 256 + threadIdx.x;
  if (i >= Bn * Nn * Dn) return;
  int d = i & 63, row = i >> 6;
  int head = d >> 4;
  const float* qp = q + row * Dn + head * 16;
  float s[Ln];
#pragma unroll
  for (int l = 0; l < Ln; ++l) {
    const float* kp = kk + (row * Ln + l) * Dn + head * 16;
    float acc = 0.f;
#pragma unroll
    for (int j = 0; j < 16; ++j) acc += qp[j] * kp[j];
    s[l] = acc * 0.25f;                  // 1/sqrt(16)
  }
  float m = fmaxf(s[0], fmaxf(s[1], s[2]));
  float e0 = __expf(s[0] - m), e1 = __expf(s[1] - m), e2 = __expf(s[2] - m);
  float inv = 1.f / (e0 + e1 + e2);
  outp[i] = e0 * inv * vv[(row * Ln + 0) * Dn + d] +
            e1 * inv * vv[(row * Ln + 1) * Dn + d] +
            e2 * inv * vv[(row * Ln + 2) * Dn + d];
}

// ---------------- GRU state blend ----------------
__global__ __launch_bounds__(256) void k_gru(float* __restrict__ h,
                                             const float* __restrict__ z,
                                             const float* __restrict__ hc,
                                             int n) {
  int i = blockIdx.x * 256 + threadIdx.x;
  if (i < n) { float zz = z[i]; h[i] = (1.f - zz) * h[i] + zz * hc[i]; }
}

// ---------------- edge decoder: tanh(mask*[h | h_nbr] @ W1 + b1) @ W2 + b2 --
__global__ __launch_bounds__(256) void k_dec(
    const float* __restrict__ h, const int* __restrict__ adj,
    const int* __restrict__ num_nodes, const float* __restrict__ W1,
    const float* __restrict__ b1, const float* __restrict__ W2,
    const float* __restrict__ b2, float* __restrict__ outw) {
  __shared__ float w1s[128 * 32];
  __shared__ float w2s[32];
  __shared__ float ys[8 * 16 * 32];
  int tid = threadIdx.x, lane = tid & 31, wave = tid >> 5;
  stage_pairs<32, 128>(W1, w1s, tid, 256);
  if (tid < 32) w2s[tid] = W2[tid];
  __syncthreads();
  int tile = blockIdx.x * 8 + wave;
  int row0 = tile * 16;
  int M = Bn * NKc;
  bool active = row0 < M;                // wave-uniform
  int half = lane >> 4, l16 = lane & 15;
  v8f acc0 = {}, acc1 = {};
  if (active) {
    int r = row0 + l16;
    int b = r / NKc;
    int rem = r - b * NKc;
    int n = rem >> 4;
    int padv = num_nodes[b];
    int av = adj[r];
    float mk = (av == padv) ? 0.f : 1.f;
    int ac = av < Nn ? av : Nn - 1;
    int src0 = (b * Nn + n) * Dn;
    int src1 = (b * Nn + ac) * Dn;
#pragma unroll 4
    for (int kt = 0; kt < 32; ++kt) {    // K = 128
      int k0 = kt * 4 + half * 2;
      int base = (k0 < 64) ? (src0 + k0) : (src1 + k0 - 64);
      v2f a = (*(const v2f*)(h + base)) * mk;
      acc0 = wmma4(a, bfrag<32>(w1s, k0, l16), acc0);
      acc1 = wmma4(a, bfrag<32>(w1s, k0, 16 + l16), acc1);
    }
    float* yw = ys + wave * 512;
#pragma unroll
    for (int v = 0; v < 8; ++v) {
      int m = half * 8 + v;
      yw[m * 32 + l16]      = tanhf(acc0[v] + b1[l16]);
      yw[m * 32 + 16 + l16] = tanhf(acc1[v] + b1[16 + l16]);
    }
  }
  asm volatile("s_wait_dscnt 0" ::: "memory");  // wave-private LDS handoff
  if (active && lane < 16) {
    const float* yw = ys + wave * 512 + l16 * 32;
    float s = b2[0];
#pragma unroll
    for (int c = 0; c < 32; ++c) s += yw[c] * w2s[c];
    outw[row0 + l16] = s;
  }
}

// ---------------- dual head: tanh(h @ W1 + b1) @ W2 + b2 -> (B,N) ----------
__global__ __launch_bounds__(256) void k_head(
    const float* __restrict__ A, const float* __restrict__ W1,
    const float* __restrict__ b1, const float* __restrict__ W2,
    const float* __restrict__ b2, float* __restrict__ outv, int M) {
  __shared__ float w1s[64 * 32];
  __shared__ float w2s[32];
  __shared__ float ys[8 * 16 * 32];
  int tid = threadIdx.x, lane = tid & 31, wave = tid >> 5;
  stage_pairs<32, 64>(W1, w1s, tid, 256);
  if (tid < 32) w2s[tid] = W2[tid];
  __syncthreads();
  int tile = blockIdx.x * 8 + wave;
  int row0 = tile * 16;
  bool active = row0 < M;
  int half = lane >> 4, l16 = lane & 15;
  v8f acc0 = {}, acc1 = {};
  if (active) {
    int arow = row0 + l16;
#pragma unroll 4
    for (int kt = 0; kt < 16; ++kt) {
      int k0 = kt * 4 + half * 2;
      v2f a = *(const v2f*)(A + arow * Dn + k0);
      acc0 = wmma4(a, bfrag<32>(w1s, k0, l16), acc0);
      acc1 = wmma4(a, bfrag<32>(w1s, k0, 16 + l16), acc1);
    }
    float* yw = ys + wave * 512;
#pragma unroll
    for (int v = 0; v < 8; ++v) {
      int m = half * 8 + v;
      yw[m * 32 + l16]      = tanhf(acc0[v] + b1[l16]);
      yw[m * 32 + 16 + l16] = tanhf(acc1[v] + b1[16 + l16]);
    }
  }
  asm volatile("s_wait_dscnt 0" ::: "memory");
  if (active && lane < 16) {
    const float* yw = ys + wave * 512 + l16 * 32;
    float s = b2[0];
#pragma unroll
    for (int c = 0; c < 32; ++c) s += yw[c] * w2s[c];
    outv[row0 + l16] = s;
  }
}

// ---------------- grouped (permuted) softmax -> dest ----------------
__global__ __launch_bounds__(256) void k_group_softmax(
    const float* __restrict__ nodew, const int* __restrict__ in_idx,
    const int* __restrict__ inv_adj, const int* __restrict__ num_nodes,
    float* __restrict__ dest) {
  int t = blockIdx.x * 256 + threadIdx.x;
  if (t >= Bn * Nn) return;
  int b = t / Nn, g = t % Nn;
  int base = b * NKc;
  int padv = num_nodes[b];
  float s[Kn];
  float mx = -1e30f;
#pragma unroll
  for (int k = 0; k < Kn; ++k) {
    int p = g * Kn + k;
    int idx = in_idx[base + p];
    float v = nodew[base + idx];
    if (inv_adj[base + p] == padv) v -= BIGC;
    s[k] = v;
    mx = fmaxf(mx, v);
  }
  float sum = 0.f;
#pragma unroll
  for (int k = 0; k < Kn; ++k) { s[k] = __expf(s[k] - mx); sum += s[k]; }
  float inv = 1.f / sum;
#pragma unroll
  for (int k = 0; k < Kn; ++k) dest[base + g * Kn + k] = s[k] * inv;
}

// ---------------- normalized weights + supply + flow init ----------------
__global__ __launch_bounds__(256) void k_normw(
    const float* __restrict__ nodew, const float* __restrict__ dest,
    const int* __restrict__ rev_idx, const int* __restrict__ adj,
    const int* __restrict__ num_nodes, const float* __restrict__ demands,
    float* __restrict__ normw, float* __restrict__ supply,
    float* __restrict__ flow0) {
  int t = blockIdx.x * 256 + threadIdx.x;
  if (t >= Bn * Nn) return;
  int b = t / Nn, n = t % Nn;
  int base = b * NKc, padv = num_nodes[b];
  float s[Kn];
  float mx = -1e30f;
#pragma unroll
  for (int k = 0; k < Kn; ++k) {
    int i = n * Kn + k;
    float nw = nodew[base + i] * dest[base + rev_idx[base + i]];
    if (adj[base + i] == padv) nw -= BIGC;
    s[k] = nw;
    mx = fmaxf(mx, nw);
  }
  float sum = 0.f;
#pragma unroll
  for (int k = 0; k < Kn; ++k) { s[k] = __expf(s[k] - mx); sum += s[k]; }
  float inv = 1.f / sum;
  float sup = fmaxf(-demands[t], 0.f);
  supply[t] = sup;
#pragma unroll
  for (int k = 0; k < Kn; ++k) {
    float w = s[k] * inv;
    normw[base + n * Kn + k] = w;
    flow0[base + n * Kn + k] = w * sup;
  }
}

// ---------------- one min-cost-flow iteration ----------------
__global__ __launch_bounds__(256) void k_flow(
    const float* __restrict__ src, const float* __restrict__ normw,
    const float* __restrict__ supply, const int* __restrict__ in_idx,
    const int* __restrict__ inv_adj, const int* __restrict__ num_nodes,
    float* __restrict__ dst) {
  int t = blockIdx.x * 256 + threadIdx.x;
  if (t >= Bn * Nn) return;
  int b = t / Nn, n = t % Nn;
  int base = b * NKc, padv = num_nodes[b];
  float infl = 0.f;
#pragma unroll
  for (int k = 0; k < Kn; ++k) {
    int i = n * Kn + k;
    if (inv_adj[base + i] != padv) infl += src[base + in_idx[base + i]];
  }
  float tot = infl + supply[t];
#pragma unroll
  for (int k = 0; k < Kn; ++k)
    dst[base + n * Kn + k] = normw[base + n * Kn + k] * tot;
}

// ---------------- flow cost + dual ascent + demand term per node ----------
__global__ __launch_bounds__(256) void k_contrib(
    const float* __restrict__ flow, const float* __restrict__ edge,
    const float* __restrict__ dvars, const int* __restrict__ adj,
    const int* __restrict__ num_nodes, const float* __restrict__ demands,
    float* __restrict__ contrib) {
  int t = blockIdx.x * 256 + threadIdx.x;
  if (t >= Bn * Nn) return;
  int b = t / Nn, n = t % Nn;
  int base = b * NKc, padv = num_nodes[b];
  float dv = dvars[t];
  float fc = 0.f, dc = 0.f;
#pragma unroll
  for (int k = 0; k < Kn; ++k) {
    int i = base + n * Kn + k;
    float e = edge[i];
    float fl = flow[i];
    fc += e * fl * fl;
    int av = adj[i];
    float am = (av == padv) ? 0.f : 1.f;
    float dtr = (av == padv) ? 0.f : dvars[b * Nn + (av < Nn ? av : Nn - 1)];
    float diff = dtr - am * dv;
    float f = 0.f, acc = 0.f;
#pragma unroll
    for (int it = 0; it < 10; ++it) {
      float grad = 2.f * e * f + diff;
      acc = 0.9f * acc + 0.01f * grad;
      f = fmaxf(f - acc, 0.f) * am;
    }
    dc += e * f * f + diff * f;
  }
  contrib[t] = fc - dc + dv * demands[t];  // flow_cost - dual_cost, per node
}

// ---------------- deterministic per-batch reduction ----------------
__global__ __launch_bounds__(256) void k_reduce(
    const float* __restrict__ contrib, float* __restrict__ outp) {
  __shared__ float sm[256];
  int b = blockIdx.x, tid = threadIdx.x;
  float s = 0.f;
  for (int i = tid; i < Nn; i += 256) s += contrib[b * Nn + i];
  sm[tid] = s;
  __syncthreads();
  for (int o = 128; o; o >>= 1) {
    if (tid < o) sm[tid] += sm[tid + o];
    __syncthreads();
  }
  if (tid == 0) outp[b] = sm[0];
}

// ---------------- host orchestration ----------------
extern "C" void kernel_launch(void* const* d_in, const int* in_sizes, int n_in,
                              void* d_out, int out_size, void* d_ws,
                              size_t ws_size, hipStream_t stream) {
  const float* demands = (const float*)d_in[0];
  const float* feats   = (const float*)d_in[1];
  const float* edge    = (const float*)d_in[2];
  const float* embt    = (const float*)d_in[3];
  const float* encW    = (const float*)d_in[4];
  const float* encB    = (const float*)d_in[5];
  const float* nbW     = (const float*)d_in[6];
  const float* Wq      = (const float*)d_in[7];
  const float* Wk      = (const float*)d_in[8];
  const float* Wv      = (const float*)d_in[9];
  const float* Wo      = (const float*)d_in[10];
  const float* gWz = (const float*)d_in[11];
  const float* gUz = (const float*)d_in[12];
  const float* gbz = (const float*)d_in[13];
  const float* gWr = (const float*)d_in[14];
  const float* gUr = (const float*)d_in[15];
  const float* gbr = (const float*)d_in[16];
  const float* gWh = (const float*)d_in[17];
  const float* gUh = (const float*)d_in[18];
  const float* gbh = (const float*)d_in[19];
  const float* decW1 = (const float*)d_in[20];
  const float* decb1 = (const float*)d_in[21];
  const float* decW2 = (const float*)d_in[22];
  const float* decb2 = (const float*)d_in[23];
  const float* duW1  = (const float*)d_in[24];
  const float* dub1  = (const float*)d_in[25];
  const float* duW2  = (const float*)d_in[26];
  const float* dub2  = (const float*)d_in[27];
  const int* adj    = (const int*)d_in[28];
  const int* invadj = (const int*)d_in[29];
  const int* neigh  = (const int*)d_in[30];
  const int* inidx  = (const int*)d_in[31];
  const int* revidx = (const int*)d_in[32];
  const int* numn   = (const int*)d_in[33];

  float* ws = (float*)d_ws;
  size_t off = 0;
  auto alloc = [&](size_t nfl) { float* p = ws + off; off += nfl; return p; };
  float* h    = alloc((size_t)Bn * Nn * Dn);
  float* agg  = alloc((size_t)Bn * Nn * Ln * Dn);
  float* st   = alloc((size_t)Bn * Nn * Ln * Dn);
  float* q    = alloc((size_t)Bn * Nn * Dn);
  float* kkb  = alloc((size_t)Bn * Nn * Ln * Dn);
  float* vvb  = alloc((size_t)Bn * Nn * Ln * Dn);
  float* at   = alloc((size_t)Bn * Nn * Dn);
  float* x    = alloc((size_t)Bn * Nn * Dn);
  float* z    = alloc((size_t)Bn * Nn * Dn);
  float* r    = alloc((size_t)Bn * Nn * Dn);
  float* hc   = alloc((size_t)Bn * Nn * Dn);
  float* nodew  = alloc((size_t)Bn * NKc);
  float* dest   = alloc((size_t)Bn * NKc);
  float* normw  = alloc((size_t)Bn * NKc);
  float* supply = alloc((size_t)Bn * Nn);
  float* fa   = alloc((size_t)Bn * NKc);
  float* fb   = alloc((size_t)Bn * NKc);
  float* dv   = alloc((size_t)Bn * Nn);
  float* ctb  = alloc((size_t)Bn * Nn);
  (void)ws_size; (void)in_sizes; (void)n_in; (void)out_size;

  const int MN = Bn * Nn;        // 20000 rows
  const int MNL = Bn * Nn * Ln;  // 60000 rows
  const int gMN = (MN / 16 + 7) / 8;
  const int gMNL = (MNL / 16 + 7) / 8;

  k_encoder<<<MN, 64, 0, stream>>>(embt, feats, encW, encB, h);

  for (int layer = 0; layer < 2; ++layer) {
    k_agg<<<MNL, 64, 0, stream>>>(h, neigh, numn, agg);
    // states = tanh(agg @ nbW)
    k_gemm64<1, false, false><<<gMNL, 256, 0, stream>>>(
        agg, nbW, nullptr, nullptr, nullptr, nullptr, st, MNL);
    // q / k / v projections
    k_gemm64<0, false, false><<<gMN, 256, 0, stream>>>(
        h, Wq, nullptr, nullptr, nullptr, nullptr, q, MN);
    k_gemm64<0, false, false><<<gMNL, 256, 0, stream>>>(
        st, Wk, nullptr, nullptr, nullptr, nullptr, kkb, MNL);
    k_gemm64<0, false, false><<<gMNL, 256, 0, stream>>>(
        st, Wv, nullptr, nullptr, nullptr, nullptr, vvb, MNL);
    k_attn<<<(MN * 64 + 255) / 256, 256, 0, stream>>>(q, kkb, vvb, at);
    // x = tanh(attn @ Wo)
    k_gemm64<1, false, false><<<gMN, 256, 0, stream>>>(
        at, Wo, nullptr, nullptr, nullptr, nullptr, x, MN);
    // GRU gates (fused dual GEMMs)
    k_gemm64<2, true, false><<<gMN, 256, 0, stream>>>(
        x, gWz, h, gUz, nullptr, gbz, z, MN);
    k_gemm64<2, true, false><<<gMN, 256, 0, stream>>>(
        x, gWr, h, gUr, nullptr, gbr, r, MN);
    k_gemm64<1, true, true><<<gMN, 256, 0, stream>>>(
        x, gWh, h, gUh, r, gbh, hc, MN);
    k_gru<<<(MN * 64 + 255) / 256, 256, 0, stream>>>(h, z, hc, MN * 64);
  }

  k_dec<<<(Bn * NKc / 16 + 7) / 8, 256, 0, stream>>>(h, adj, numn, decW1, decb1,
                                                     decW2, decb2, nodew);
  k_head<<<gMN, 256, 0, stream>>>(h, duW1, dub1, duW2, dub2, dv, MN);

  k_group_softmax<<<(MN + 255) / 256, 256, 0, stream>>>(nodew, inidx, invadj,
                                                        numn, dest);
  k_normw<<<(MN + 255) / 256, 256, 0, stream>>>(nodew, dest, revidx, adj, numn,
                                                demands, normw, supply, fa);
  for (int it = 0; it < 10; ++it) {
    const float* s = (it & 1) ? fb : fa;
    float* d2 = (it & 1) ? fa : fb;
    k_flow<<<(MN + 255) / 256, 256, 0, stream>>>(s, normw, supply, inidx,
                                                 invadj, numn, d2);
  }
  // 10 iterations: final flow lands in fa
  k_contrib<<<(MN + 255) / 256, 256, 0, stream>>>(fa, edge, dv, adj, numn,
                                                  demands, ctb);
  k_reduce<<<Bn, 256, 0, stream>>>(ctb, (float*)d_out);
}